// ResnetBlockGroupNormShallowConv1d_1580547974682
// MI455X (gfx1250) — compile-verified
//
#include <hip/hip_runtime.h>

// ResnetBlock: GroupNorm(channels-only stats, ddof=1) -> ReLU -> grouped 1x1 conv -> residual
// x:(8,256,32768) f32, gamma/beta:(256), w:(256,32). out = x + conv(relu(gn(x))).
//
// Memory-bound: ~0.5 GB traffic -> ~22us @ 23.3 TB/s; conv math (4.3 GFLOP f32) runs on
// V_WMMA_F32_16X16X4_F32. Wave w owns N-tile w; computes both M-tiles with compile-time
// indices so the A-operand register array is statically addressed (no cndmask trees).

typedef __attribute__((ext_vector_type(2))) float v2f;
typedef __attribute__((ext_vector_type(4))) float v4f;
typedef __attribute__((ext_vector_type(8))) float v8f;

#define NB   8
#define ND   256
#define NT   32768
#define NG   8       // conv groups
#define CPG  32      // channels per conv group
#define TT   128     // t-tile per block
#define PAD  4
#define STR  (TT + PAD)
#define EPSV 1e-5f

__global__ __launch_bounds__(256) void resblock_gn_conv1x1_kernel(
    const float* __restrict__ x, const float* __restrict__ gamma,
    const float* __restrict__ beta, const float* __restrict__ w,
    float* __restrict__ out)
{
    __shared__ __align__(16) float xs[CPG][STR];   // x tile (becomes output tile)
    __shared__ __align__(16) float hs[CPG][STR];   // relu(gn(x)) tile
    __shared__ float gammas[CPG], betas[CPG];

    const int tid  = threadIdx.x;
    const int lane = tid & 31;
    const int wave = tid >> 5;

    // block -> (b, g, t-slab);  T/TT = 256 slabs
    const int  bid = blockIdx.x;
    const int  ts  = bid & 255;
    const int  g   = (bid >> 8) & 7;
    const int  b   = bid >> 11;
    const long xbase = ((long)b * ND + (long)g * CPG) * NT + (long)ts * TT;

    // ---------- phase 1: coalesced load x tile -> LDS ----------
    // 32 rows x 128 floats = 1024 float4s; 256 threads x 4 reps
    #pragma unroll
    for (int rep = 0; rep < 4; ++rep) {
        int i   = tid + rep * 256;
        int row = i >> 5;                // TT/4 = 32 float4 per row
        int col = (i & 31) * 4;
        v4f v = *(const v4f*)(x + xbase + (long)row * NT + col);
        *(v4f*)&xs[row][col] = v;
    }
    if (tid < CPG) {
        gammas[tid] = gamma[g * CPG + tid];
        betas[tid]  = beta[g * CPG + tid];
    }

    // ---------- preload W_g into WMMA A-operand registers ----------
    // 32-bit A 16x4 layout: lane holds A[M = lane&15][K = kbase + 2*(lane>>4) + v], v=0,1
    // All indices below are compile-time after unrolling -> pure register file.
    float areg[2][8][2];
    {
        const int m    = lane & 15;
        const int koff = (lane >> 4) << 1;
        #pragma unroll
        for (int mt = 0; mt < 2; ++mt)
            #pragma unroll
            for (int k = 0; k < 8; ++k) {
                const float* wp = w + (long)(g * CPG + mt * 16 + m) * CPG + k * 4 + koff;
                areg[mt][k][0] = wp[0];
                areg[mt][k][1] = wp[1];
            }
    }

    __syncthreads();

    // ---------- phase 2: GroupNorm (8-channel stats per column) + ReLU ----------
    // 128 columns, 4 sub-groups of 8 channels; thread = (column, pair-of-subgroups)
    {
        const int c   = tid & 127;
        const int sgp = tid >> 7;        // 0 or 1 -> subgroups {0,1} or {2,3}
        #pragma unroll
        for (int s = 0; s < 2; ++s) {
            const int sg = sgp * 2 + s;
            float v[8], sum = 0.f, sumsq = 0.f;
            #pragma unroll
            for (int j = 0; j < 8; ++j) {
                float t = xs[sg * 8 + j][c];
                v[j] = t; sum += t; sumsq += t * t;
            }
            float mean = sum * 0.125f;
            float var  = (sumsq - sum * mean) * (1.0f / 7.0f);   // ddof=1
            float inv  = rsqrtf(var + EPSV);
            #pragma unroll
            for (int j = 0; j < 8; ++j) {
                int ch = sg * 8 + j;
                float h = (v[j] - mean) * inv * gammas[ch] + betas[ch];
                hs[ch][c] = fmaxf(h, 0.f);
            }
        }
    }

    __syncthreads();

    // ---------- phase 3: WMMA  dx = W_g @ h, residual in-place ----------
    // wave w -> N-tile nt = w (col offset only); both M-tiles computed here with
    // compile-time mt. B operand shared between the two WMMAs per K-step.
    {
        const int m15  = lane & 15;
        const int krow = (lane >> 4) << 1;   // B: VGPR0 = K{0,2}, VGPR1 = K{1,3}
        const int col  = wave * 16 + m15;    // nt = wave, 8 waves x 16 = 128 cols
        v8f acc0 = {}, acc1 = {};
        #pragma unroll
        for (int k = 0; k < 8; ++k) {
            v2f bv;
            bv.x = hs[k * 4 + krow + 0][col];
            bv.y = hs[k * 4 + krow + 1][col];
            v2f a0; a0.x = areg[0][k][0]; a0.y = areg[0][k][1];
            v2f a1; a1.x = areg[1][k][0]; a1.y = areg[1][k][1];
            acc0 = __builtin_amdgcn_wmma_f32_16x16x4_f32(
                    false, a0, false, bv, (short)0, acc0, false, false);
            acc1 = __builtin_amdgcn_wmma_f32_16x16x4_f32(
                    false, a1, false, bv, (short)0, acc1, false, false);
        }
        // C/D layout: VGPR r -> M = r (lanes 0-15) or M = 8+r (lanes 16-31)
        const int rhalf = (lane >> 4) << 3;
        #pragma unroll
        for (int r = 0; r < 8; ++r) {
            xs[r + rhalf][col]      += acc0[r];
            xs[16 + r + rhalf][col] += acc1[r];
        }
    }

    __syncthreads();

    // ---------- phase 4: coalesced store out = xs ----------
    #pragma unroll
    for (int rep = 0; rep < 4; ++rep) {
        int i   = tid + rep * 256;
        int row = i >> 5;
        int col = (i & 31) * 4;
        v4f v = *(const v4f*)&xs[row][col];
        *(v4f*)(out + xbase + (long)row * NT + col) = v;
    }
}

extern "C" void kernel_launch(void* const* d_in, const int* in_sizes, int n_in,
                              void* d_out, int out_size, void* d_ws, size_t ws_size,
                              hipStream_t stream) {
    const float* x     = (const float*)d_in[0];
    const float* gamma = (const float*)d_in[1];
    const float* beta  = (const float*)d_in[2];
    const float* w     = (const float*)d_in[3];
    float* out = (float*)d_out;

    const int blocks = NB * NG * (NT / TT);   // 8*8*256 = 16384
    resblock_gn_conv1x1_kernel<<<blocks, 256, 0, stream>>>(x, gamma, beta, w, out);
}